// SelectiveSSM_81724637708852
// MI455X (gfx1250) — compile-verified
//
#include <hip/hip_runtime.h>
#include <hip/hip_bf16.h>
#include <math.h>

#define D_INNER 2048
#define D_STATE 16
#define DT_RANK 128
#define BATCH   4
#define SEQLEN  4096
#define E_DIM   (DT_RANK + 2 * D_STATE)   // 160
#define M_ROWS  (BATCH * SEQLEN)          // 16384

typedef __attribute__((ext_vector_type(2))) float v2f;
typedef __attribute__((ext_vector_type(8))) float v8f;

// ---------------------------------------------------------------------------
// Kernel 1: x_dbl[M,160] = x[M,2048] @ x_proj_w[160,2048]^T  (fp32 WMMA)
// One wave per 16x16 output tile. 512 k-steps of 4, dual accumulators.
// ---------------------------------------------------------------------------
__global__ __launch_bounds__(128) void k1_xproj(const float* __restrict__ X,
                                                const float* __restrict__ W,
                                                float* __restrict__ out) {
    const int wave = blockIdx.x * (blockDim.x >> 5) + (threadIdx.x >> 5);
    const int lane = threadIdx.x & 31;
    const int NT = E_DIM / 16;            // 10
    const int tm = wave / NT;
    const int tn = wave % NT;
    const int r    = lane & 15;
    const int half = lane >> 4;

    const float* xr = X + (size_t)(tm * 16 + r) * D_INNER;  // A: row = lane&15
    const float* wr = W + (size_t)(tn * 16 + r) * D_INNER;  // B: col = lane&15 (W row-major over K)

    v8f acc0 = {0.f, 0.f, 0.f, 0.f, 0.f, 0.f, 0.f, 0.f};
    v8f acc1 = {0.f, 0.f, 0.f, 0.f, 0.f, 0.f, 0.f, 0.f};

    for (int k = 0; k < D_INNER; k += 8) {
        const int ko0 = k + 2 * half;
        const int ko1 = k + 4 + 2 * half;
        v2f a0 = *(const v2f*)(xr + ko0);
        v2f b0 = *(const v2f*)(wr + ko0);
        v2f a1 = *(const v2f*)(xr + ko1);
        v2f b1 = *(const v2f*)(wr + ko1);
        acc0 = __builtin_amdgcn_wmma_f32_16x16x4_f32(false, a0, false, b0,
                                                     (short)0, acc0, false, false);
        acc1 = __builtin_amdgcn_wmma_f32_16x16x4_f32(false, a1, false, b1,
                                                     (short)0, acc1, false, false);
    }

    // C/D layout: VGPR i -> (M = tm*16 + i + 8*half, N = tn*16 + (lane&15))
    const int col = tn * 16 + r;
#pragma unroll
    for (int i = 0; i < 8; ++i) {
        float v = acc0[i] + acc1[i];
        out[(size_t)(tm * 16 + i + 8 * half) * E_DIM + col] = v;
    }
}

// ---------------------------------------------------------------------------
// Kernel 2: delta[M,2048] = softplus(x_dbl[:, :128] @ dt_proj_w[2048,128]^T + b)
// ---------------------------------------------------------------------------
__global__ __launch_bounds__(128) void k2_dtproj(const float* __restrict__ XD,   // (M,160)
                                                 const float* __restrict__ W,    // (2048,128)
                                                 const float* __restrict__ bias, // (2048,)
                                                 float* __restrict__ delta) {    // (M,2048)
    const int wave = blockIdx.x * (blockDim.x >> 5) + (threadIdx.x >> 5);
    const int lane = threadIdx.x & 31;
    const int NT = D_INNER / 16;          // 128
    const int tm = wave / NT;
    const int tn = wave % NT;
    const int r    = lane & 15;
    const int half = lane >> 4;

    const float* xr = XD + (size_t)(tm * 16 + r) * E_DIM;
    const float* wr = W  + (size_t)(tn * 16 + r) * DT_RANK;

    v8f acc0 = {0.f, 0.f, 0.f, 0.f, 0.f, 0.f, 0.f, 0.f};
    v8f acc1 = {0.f, 0.f, 0.f, 0.f, 0.f, 0.f, 0.f, 0.f};

    for (int k = 0; k < DT_RANK; k += 8) {
        const int ko0 = k + 2 * half;
        const int ko1 = k + 4 + 2 * half;
        v2f a0 = *(const v2f*)(xr + ko0);
        v2f b0 = *(const v2f*)(wr + ko0);
        v2f a1 = *(const v2f*)(xr + ko1);
        v2f b1 = *(const v2f*)(wr + ko1);
        acc0 = __builtin_amdgcn_wmma_f32_16x16x4_f32(false, a0, false, b0,
                                                     (short)0, acc0, false, false);
        acc1 = __builtin_amdgcn_wmma_f32_16x16x4_f32(false, a1, false, b1,
                                                     (short)0, acc1, false, false);
    }

    const int col = tn * 16 + r;
    const float bv = bias[col];
#pragma unroll
    for (int i = 0; i < 8; ++i) {
        float z = acc0[i] + acc1[i] + bv;
        float sp = (z > 20.f) ? z : log1pf(__expf(z));  // softplus
        delta[(size_t)(tm * 16 + i + 8 * half) * D_INNER + col] = sp;
    }
}

// ---------------------------------------------------------------------------
// Kernel 3: selective scan. One lane per (b,d) channel; h[16] in VGPRs.
//   h[n] = exp(dv*A[d,n]) * h[n] + (dv*xv) * B_t[n];  y = sum h[n]*C_t[n] + D*xv
// B_t/C_t live at x_dbl[b,l,128:160] -> wave-uniform scalar loads.
// ---------------------------------------------------------------------------
__global__ __launch_bounds__(256) void k3_scan(const float* __restrict__ x,
                                               const float* __restrict__ delta,
                                               const float* __restrict__ xdbl,
                                               const float* __restrict__ A_log,
                                               const float* __restrict__ Dp,
                                               float* __restrict__ y) {
    const int dblk = D_INNER / 256;                        // 8 blocks per batch
    const int b = blockIdx.x / dblk;
    const int d = (blockIdx.x % dblk) * 256 + threadIdx.x;

    float A[D_STATE];
#pragma unroll
    for (int n = 0; n < D_STATE; ++n)
        A[n] = -__expf(A_log[(size_t)d * D_STATE + n]);
    const float Dv = Dp[d];

    float h[D_STATE];
#pragma unroll
    for (int n = 0; n < D_STATE; ++n) h[n] = 0.f;

    const float* xp  = x     + (size_t)b * SEQLEN * D_INNER + d;
    const float* dp  = delta + (size_t)b * SEQLEN * D_INNER + d;
    const float* bcp = xdbl  + (size_t)b * SEQLEN * E_DIM + DT_RANK;
    float*       yp  = y     + (size_t)b * SEQLEN * D_INNER + d;

    for (int l = 0; l < SEQLEN; ++l) {
        const float xv = xp[(size_t)l * D_INNER];
        const float dv = dp[(size_t)l * D_INNER];
        const float* bc = bcp + (size_t)l * E_DIM;   // uniform across the wave
        const float sx = dv * xv;
        float acc = Dv * xv;
#pragma unroll
        for (int n = 0; n < D_STATE; ++n) {
            const float dA = __expf(dv * A[n]);      // v_exp_f32
            h[n] = dA * h[n] + sx * bc[n];           // B_t[n]
            acc += h[n] * bc[D_STATE + n];           // C_t[n]
        }
        yp[(size_t)l * D_INNER] = acc;
    }
}

// ---------------------------------------------------------------------------
extern "C" void kernel_launch(void* const* d_in, const int* in_sizes, int n_in,
                              void* d_out, int out_size, void* d_ws, size_t ws_size,
                              hipStream_t stream) {
    const float* x         = (const float*)d_in[0];  // (4,4096,2048)
    const float* A_log     = (const float*)d_in[1];  // (2048,16)
    const float* D_param   = (const float*)d_in[2];  // (2048,)
    const float* x_proj_w  = (const float*)d_in[3];  // (160,2048)
    const float* dt_proj_w = (const float*)d_in[4];  // (2048,128)
    const float* dt_proj_b = (const float*)d_in[5];  // (2048,)
    float* out = (float*)d_out;

    char* ws = (char*)d_ws;
    float* xdbl  = (float*)ws;                                            // 16384*160*4  = 10.5 MB
    float* delta = (float*)(ws + (size_t)M_ROWS * E_DIM * sizeof(float)); // 16384*2048*4 = 134 MB

    // K1: 1024 m-tiles * 10 n-tiles = 10240 waves, 4 waves/block
    {
        const int waves = (M_ROWS / 16) * (E_DIM / 16);
        k1_xproj<<<dim3(waves / 4), dim3(128), 0, stream>>>(x, x_proj_w, xdbl);
    }
    // K2: 1024 * 128 = 131072 waves, 4 waves/block
    {
        const int waves = (M_ROWS / 16) * (D_INNER / 16);
        k2_dtproj<<<dim3(waves / 4), dim3(128), 0, stream>>>(xdbl, dt_proj_w, dt_proj_b, delta);
    }
    // K3: 8192 channels -> 32 blocks of 256
    {
        k3_scan<<<dim3(BATCH * (D_INNER / 256)), dim3(256), 0, stream>>>(
            x, delta, xdbl, A_log, D_param, out);
    }
}